// TextProposal_68599217651819
// MI455X (gfx1250) — compile-verified
//
#include <hip/hip_runtime.h>
#include <stdint.h>

#define B_      4
#define N_      20000
#define V_      12000
#define MAXOUT_ 2000
#define IOU_T   0.3f
#define SCORE_T 0.7f

typedef unsigned int u32x4 __attribute__((ext_vector_type(4)));
typedef int          i32x8 __attribute__((ext_vector_type(8)));
typedef int          i32x4 __attribute__((ext_vector_type(4)));

// ---------------------------------------------------------------------------
// Kernel 1: gather valid anchors, softmax fg score, box regression
// ---------------------------------------------------------------------------
__global__ __launch_bounds__(256) void prep_kernel(
    const float* __restrict__ deltas,   // (B, N, 2)
    const float* __restrict__ logits,   // (B, N, 2)
    const float* __restrict__ anchors,  // (V, 4)
    const int*   __restrict__ vidx,     // (V,)
    float* __restrict__ score, float* __restrict__ px1, float* __restrict__ py1,
    float* __restrict__ px2,   float* __restrict__ py2,
    float* __restrict__ pl0,   float* __restrict__ pl1) {
  int idx = blockIdx.x * blockDim.x + threadIdx.x;
  if (idx >= B_ * V_) return;
  int b = idx / V_;
  int i = idx - b * V_;
  int a = vidx[i];

  size_t base = ((size_t)b * N_ + (size_t)a) * 2;
  float d0 = deltas[base + 0];
  float d1 = deltas[base + 1];
  float l0 = logits[base + 0];
  float l1 = logits[base + 1];

  // softmax over 2 classes, fg prob = class 1
  float s = 1.0f / (1.0f + expf(l0 - l1));

  float x1a = anchors[i * 4 + 0];
  float y1a = anchors[i * 4 + 1];
  float x2a = anchors[i * 4 + 2];
  float y2a = anchors[i * 4 + 3];

  float h  = y2a - y1a;
  float cy = (y1a + y2a) * 0.5f;
  float dy = d0 * 0.1f;
  float dh = d1 * 0.2f;
  cy = cy + dy * h;
  h  = h * expf(dh);

  score[idx] = s;
  px1[idx] = x1a;                 // USE_SIDE_REFINE == False
  py1[idx] = cy - 0.5f * h;
  px2[idx] = x2a;
  py2[idx] = cy + 0.5f * h;
  pl0[idx] = l0;
  pl1[idx] = l1;
}

// ---------------------------------------------------------------------------
// Kernel 2: stable descending rank via LDS-tiled counting; scatter permutation.
// rank(i) = #{j : s_j > s_i  or  (s_j == s_i and j < i)}   (== stable argsort(-s))
// ---------------------------------------------------------------------------
__global__ __launch_bounds__(256) void rank_kernel(const float* __restrict__ score,
                                                   int* __restrict__ ord) {
  const int b = blockIdx.y;
  const int i = blockIdx.x * blockDim.x + threadIdx.x;
  __shared__ float tile[256];
  float s = (i < V_) ? score[(size_t)b * V_ + i] : 0.0f;
  int rank = 0;
  for (int basej = 0; basej < V_; basej += 256) {
    int j = basej + threadIdx.x;
    tile[threadIdx.x] = (j < V_) ? score[(size_t)b * V_ + j] : -1e30f;
    __syncthreads();
    int lim = V_ - basej; if (lim > 256) lim = 256;
    if (i < V_) {
      for (int t = 0; t < lim; ++t) {
        float sj = tile[t];
        int jj = basej + t;
        rank += (sj > s) || (sj == s && jj < i);
      }
    }
    __syncthreads();
  }
  if (i < V_) ord[(size_t)b * V_ + rank] = i;   // permutation => race-free scatter
}

// ---------------------------------------------------------------------------
// Kernel 3: gather into sorted SoA. sboxes layout per batch: [x1|y1|x2|y2][V]
// (contiguous 192 KB tile per batch -> one TDM descriptor)
// ---------------------------------------------------------------------------
__global__ __launch_bounds__(256) void gather_kernel(
    const int* __restrict__ ord, const float* __restrict__ score,
    const float* __restrict__ px1, const float* __restrict__ py1,
    const float* __restrict__ px2, const float* __restrict__ py2,
    const float* __restrict__ pl0, const float* __restrict__ pl1,
    float* __restrict__ sboxes, float* __restrict__ sscore,
    float* __restrict__ sl0, float* __restrict__ sl1) {
  int idx = blockIdx.x * blockDim.x + threadIdx.x;
  if (idx >= B_ * V_) return;
  int b = idx / V_;
  int r = idx - b * V_;
  int i = ord[idx];
  size_t src = (size_t)b * V_ + i;
  float* sb = sboxes + (size_t)b * 4 * V_;
  sb[0 * V_ + r] = px1[src];
  sb[1 * V_ + r] = py1[src];
  sb[2 * V_ + r] = px2[src];
  sb[3 * V_ + r] = py2[src];
  sscore[idx] = score[src];
  sl0[idx]    = pl0[src];
  sl1[idx]    = pl1[src];
}

// ---------------------------------------------------------------------------
// Kernel 4: one block per batch. TDM-load sorted boxes (192 KB) into LDS,
// greedy NMS on an LDS keep-bitmask, popcount-rank compaction + padded writes.
// ---------------------------------------------------------------------------
__global__ __launch_bounds__(1024) void nms_kernel(
    const float* __restrict__ sboxes, const float* __restrict__ sscore,
    const float* __restrict__ sl0, const float* __restrict__ sl1,
    float* __restrict__ out) {
  const int b   = blockIdx.x;
  const int tid = threadIdx.x;
  const int T   = blockDim.x;

  __shared__ float    sb[4 * V_];          // 192000 B: x1 | y1 | x2 | y2
  __shared__ unsigned keep[V_ / 32];       // 375 dwords

  const float* gsb = sboxes + (size_t)b * 4 * V_;

#if __has_builtin(__builtin_amdgcn_tensor_load_to_lds)
  // --- Tensor Data Mover: DMA 4*V_ dwords global -> LDS (wave 0 issues) ---
  if (tid < 32) {
    uint64_t ga = (uint64_t)(uintptr_t)gsb;
    unsigned la = (unsigned)(uintptr_t)(void*)&sb[0];   // low 32 bits = LDS offset
    u32x4 g0;
    g0[0] = 1u;                                          // count=1, user D#
    g0[1] = la;                                          // lds_addr (bytes)
    g0[2] = (unsigned)(ga & 0xFFFFFFFFull);              // global_addr[31:0]
    g0[3] = (unsigned)((ga >> 32) & 0x1FFFFFFull)        // global_addr[56:32]
            | (2u << 30);                                // type = 2 ("image")
    i32x8 g1;
    g1[0] = (int)(2u << 16);          // data_size = 4 B; no multicast/pad/iterate
    g1[1] = (int)(48000u << 16);      // tensor_dim0 lo16 = 48000
    g1[2] = (int)(1u << 16);          // tensor_dim0 hi16 = 0; tensor_dim1 lo16 = 1
    g1[3] = (int)(48000u << 16);      // tensor_dim1 hi16 = 0; tile_dim0 = 48000
    g1[4] = 1;                        // tile_dim1 = 1; tile_dim2 = 0 (unused)
    g1[5] = 48000;                    // tensor_dim0_stride lo32
    g1[6] = (int)(0xBB80u << 16);     // stride0 hi16 = 0; tensor_dim1_stride lo16
    g1[7] = 0;                        // tensor_dim1_stride hi32
    i32x4 g2 = {0, 0, 0, 0};          // dims 2..4 unused (tile_dim3/4 = 0)
    i32x4 g3 = {0, 0, 0, 0};
    i32x8 g4 = {0, 0, 0, 0, 0, 0, 0, 0};  // extra operand on clang-23 lane (zero)
    // amdgpu-toolchain (clang-23) 6-arg form:
    // (uint32x4, int32x8, int32x4, int32x4, int32x8, i32 cpol)
    __builtin_amdgcn_tensor_load_to_lds(g0, g1, g2, g3, g4, 0);
    __builtin_amdgcn_s_wait_tensorcnt(0);
  }
#else
  for (int k = tid; k < 4 * V_; k += T) sb[k] = gsb[k];
#endif

  // keep-bitmask init (score > thresh) and zero-fill this batch's output rows
  for (int w = tid; w < V_ / 32; w += T) {
    unsigned m = 0;
    for (int t = 0; t < 32; ++t)
      m |= (sscore[(size_t)b * V_ + w * 32 + t] > SCORE_T) ? (1u << t) : 0u;
    keep[w] = m;
  }
  float* ob = out + (size_t)b * MAXOUT_ * 5;
  float* os = out + (size_t)B_ * MAXOUT_ * 5 + (size_t)b * MAXOUT_ * 2;
  float* ol = out + (size_t)B_ * MAXOUT_ * 7 + (size_t)b * MAXOUT_ * 3;
  for (int k = tid; k < MAXOUT_ * 5; k += T) ob[k] = 0.0f;
  for (int k = tid; k < MAXOUT_ * 2; k += T) os[k] = 0.0f;
  for (int k = tid; k < MAXOUT_ * 3; k += T) ol[k] = 0.0f;
  __syncthreads();

  const float* lx1 = sb;
  const float* ly1 = sb + V_;
  const float* lx2 = sb + 2 * V_;
  const float* ly2 = sb + 3 * V_;

  // greedy NMS: suppress j > i iff keep[i] and IoU > thresh (reference semantics)
  for (int i = 0; i < V_; ++i) {
    if ((keep[i >> 5] >> (i & 31)) & 1u) {
      float xi1 = lx1[i], yi1 = ly1[i], xi2 = lx2[i], yi2 = ly2[i];
      float ai = (xi2 - xi1) * (yi2 - yi1);
      for (int j = i + 1 + tid; j < V_; j += T) {
        if (!((keep[j >> 5] >> (j & 31)) & 1u)) continue;   // monotonic: safe skip
        float xx1 = fmaxf(xi1, lx1[j]);
        float yy1 = fmaxf(yi1, ly1[j]);
        float xx2 = fminf(xi2, lx2[j]);
        float yy2 = fminf(yi2, ly2[j]);
        float inter = fmaxf(xx2 - xx1, 0.0f) * fmaxf(yy2 - yy1, 0.0f);
        float aj = (lx2[j] - lx1[j]) * (ly2[j] - ly1[j]);
        float iou = inter / (ai + aj - inter + 1e-10f);
        if (iou > IOU_T) atomicAnd(&keep[j >> 5], ~(1u << (j & 31)));
      }
    }
    __syncthreads();
  }

  // compaction: rank = popcount of kept bits before r; write rows with tag=1
  for (int r = tid; r < V_; r += T) {
    if (!((keep[r >> 5] >> (r & 31)) & 1u)) continue;
    int w = r >> 5;
    int rank = 0;
    for (int t = 0; t < w; ++t) rank += __popc(keep[t]);
    rank += __popc(keep[w] & ((1u << (r & 31)) - 1u));
    if (rank < MAXOUT_) {
      ob[rank * 5 + 0] = lx1[r];
      ob[rank * 5 + 1] = ly1[r];
      ob[rank * 5 + 2] = lx2[r];
      ob[rank * 5 + 3] = ly2[r];
      ob[rank * 5 + 4] = 1.0f;
      os[rank * 2 + 0] = sscore[(size_t)b * V_ + r];
      os[rank * 2 + 1] = 1.0f;
      ol[rank * 3 + 0] = sl0[(size_t)b * V_ + r];
      ol[rank * 3 + 1] = sl1[(size_t)b * V_ + r];
      ol[rank * 3 + 2] = 1.0f;
    }
  }
}

// ---------------------------------------------------------------------------
extern "C" void kernel_launch(void* const* d_in, const int* in_sizes, int n_in,
                              void* d_out, int out_size, void* d_ws, size_t ws_size,
                              hipStream_t stream) {
  const float* deltas  = (const float*)d_in[0];
  // d_in[1] = side_deltas: unused (USE_SIDE_REFINE == False => dx never read)
  const float* logits  = (const float*)d_in[2];
  const float* anchors = (const float*)d_in[3];
  const int*   vidx    = (const int*)  d_in[4];
  float*       out     = (float*)d_out;

  const size_t BV = (size_t)B_ * V_;
  float* ws     = (float*)d_ws;
  float* score  = ws;               // BV
  float* px1    = score + BV;       // BV
  float* py1    = px1   + BV;       // BV
  float* px2    = py1   + BV;       // BV
  float* py2    = px2   + BV;       // BV
  float* pl0    = py2   + BV;       // BV
  float* pl1    = pl0   + BV;       // BV
  int*   ord    = (int*)(pl1 + BV); // BV ints
  float* sboxes = (float*)(ord + BV);   // 4*BV  [b][comp][r]
  float* sscore = sboxes + 4 * BV;      // BV
  float* sl0    = sscore + BV;          // BV
  float* sl1    = sl0    + BV;          // BV
  (void)in_sizes; (void)n_in; (void)out_size; (void)ws_size;

  const int blocks = (int)((BV + 255) / 256);
  prep_kernel<<<blocks, 256, 0, stream>>>(deltas, logits, anchors, vidx,
                                          score, px1, py1, px2, py2, pl0, pl1);
  rank_kernel<<<dim3((V_ + 255) / 256, B_), 256, 0, stream>>>(score, ord);
  gather_kernel<<<blocks, 256, 0, stream>>>(ord, score, px1, py1, px2, py2,
                                            pl0, pl1, sboxes, sscore, sl0, sl1);
  nms_kernel<<<B_, 1024, 0, stream>>>(sboxes, sscore, sl0, sl1, out);
}